// FullCapsuleLayer_42545946035038
// MI455X (gfx1250) — compile-verified
//
#include <hip/hip_runtime.h>
#include <hip/hip_bf16.h>

typedef __attribute__((ext_vector_type(2))) float v2f;
typedef __attribute__((ext_vector_type(8))) float v8f;

#define BATCH 32
#define NN    3200          // H*W*IC
#define IA    32            // input atoms
#define CAPS  20            // NUM_CAPSULE
#define ATOMS 64            // NUM_ATOMS
#define CA    (CAPS*ATOMS)  // 1280
#define KTOT  (NN*IA)       // 102400
#define KCH   1024          // K per workgroup chunk in pass A (32 n-values)
#define VSTR  1288          // padded LDS row stride for votes (bank shift)

// -------------------------------------------------------------------------
// Pass A: preact0_raw[b,ca] = sum_{n,ia} x[b,n,ia] * W[n,ia,ca]
// (uniform route 1/C folded into the squash epilogue)
// Plain split-K GEMM [32 x 1280 x 102400] with f32 WMMA 16x16x4.
// Grid: KTOT/KCH = 100 workgroups x 512 threads (16 waves); each wave owns a
// 32x80 output slice (2 M-tiles x 5 N-tiles), atomically reduced at the end.
// -------------------------------------------------------------------------
__global__ __launch_bounds__(512) void capsule_votes_sum_kernel(
    const float* __restrict__ x, const float* __restrict__ W,
    float* __restrict__ preact)
{
    extern __shared__ float lds[];              // x chunk: [32][KCH] = 128 KB
    const int t    = threadIdx.x;
    const int lane = t & 31;
    const int wv   = t >> 5;                    // 0..15
    const long kbase = (long)blockIdx.x * KCH;

    // Stage this K-chunk of x into LDS (x[b] rows are contiguous length KTOT)
    for (int i = t; i < BATCH * KCH; i += 512) {
        const int b = i / KCH, k = i % KCH;
        lds[i] = x[(long)b * KTOT + kbase + k];
    }
    __syncthreads();

    const int col0    = wv * 80;
    const int krow_lo = (lane >> 4) << 1;       // 0 for lanes 0-15, 2 for 16-31
    const int ncol    = lane & 15;
    const int mrow    = lane & 15;

    v8f acc[2][5] = {};
    for (int k0 = 0; k0 < KCH; k0 += 4) {
        // A fragments (from LDS): a.x = X[m][k0+krow_lo], a.y = next K
        v2f a0, a1;
        {
            const float* p0 = &lds[mrow * KCH + k0 + krow_lo];
            const float* p1 = &lds[(16 + mrow) * KCH + k0 + krow_lo];
            a0.x = p0[0]; a0.y = p0[1];
            a1.x = p1[0]; a1.y = p1[1];
        }
        const long kg = kbase + k0 + krow_lo;
        const float* wr0 = W + kg * CA;         // B row K
        const float* wr1 = wr0 + CA;            // B row K+1
        __builtin_prefetch(wr0 + 16 * CA + col0 + ncol, 0, 1); // stream-ahead
#pragma unroll
        for (int nt = 0; nt < 5; ++nt) {
            const int cc = col0 + nt * 16 + ncol;
            v2f bf; bf.x = wr0[cc]; bf.y = wr1[cc];
            acc[0][nt] = __builtin_amdgcn_wmma_f32_16x16x4_f32(
                false, a0, false, bf, (short)0, acc[0][nt], false, false);
            acc[1][nt] = __builtin_amdgcn_wmma_f32_16x16x4_f32(
                false, a1, false, bf, (short)0, acc[1][nt], false, false);
        }
    }

    // Split-K reduction into L2-resident preact[32][1280]
#pragma unroll
    for (int mt = 0; mt < 2; ++mt)
#pragma unroll
        for (int nt = 0; nt < 5; ++nt)
#pragma unroll
            for (int v = 0; v < 8; ++v) {
                const int row = mt * 16 + v + ((lane >> 4) << 3);
                const int cc  = col0 + nt * 16 + ncol;
                atomicAdd(&preact[row * CA + cc], acc[mt][nt][v]);
            }
}

// -------------------------------------------------------------------------
// Pass B (iterations 1,2): per n, compute votes_n = x_n(32x32) @ W_n(32x1280)
// via WMMA into LDS, then logits update + softmax + route-weighted preact
// accumulation. Each workgroup loops over 16 n-values, keeping per-thread
// preact partials in registers; one atomic flush at the end.
// -------------------------------------------------------------------------
__global__ __launch_bounds__(512) void capsule_routing_pass_kernel(
    const float* __restrict__ x, const float* __restrict__ W,
    float* __restrict__ logits, const float* __restrict__ act,
    float* __restrict__ preact, int first)
{
    extern __shared__ float lds[];
    float* votes = lds;                      // [32][VSTR] padded
    float* xs    = lds + BATCH * VSTR;       // [32][32]
    float* larr  = xs + BATCH * IA;          // [32][20] logits -> route

    const int t    = threadIdx.x;
    const int lane = t & 31;
    const int wv   = t >> 5;
    const int col0    = wv * 80;
    const int krow_lo = (lane >> 4) << 1;
    const int ncol    = lane & 15;
    const int mrow    = lane & 15;

    float pacc[80];
#pragma unroll
    for (int j = 0; j < 80; ++j) pacc[j] = 0.f;

    for (int n = blockIdx.x; n < NN; n += gridDim.x) {
        __syncthreads();                                 // protect prev-iter reads
        for (int i = t; i < BATCH * IA; i += 512) {
            const int b = i >> 5, ia = i & 31;
            xs[i] = x[((long)b * NN + n) * IA + ia];
        }
        __syncthreads();

        // ---- votes GEMM: 32x1280, K = 32 -> 8 WMMA k-steps per wave slice
        const float* Wn = W + (long)n * IA * CA;
        v8f acc[2][5] = {};
#pragma unroll
        for (int k0 = 0; k0 < IA; k0 += 4) {
            v2f a0, a1;
            {
                const float* p0 = &xs[mrow * IA + k0 + krow_lo];
                const float* p1 = &xs[(16 + mrow) * IA + k0 + krow_lo];
                a0.x = p0[0]; a0.y = p0[1];
                a1.x = p1[0]; a1.y = p1[1];
            }
            const float* wr0 = Wn + (long)(k0 + krow_lo) * CA;
            const float* wr1 = wr0 + CA;
#pragma unroll
            for (int nt = 0; nt < 5; ++nt) {
                const int cc = col0 + nt * 16 + ncol;
                v2f bf; bf.x = wr0[cc]; bf.y = wr1[cc];
                acc[0][nt] = __builtin_amdgcn_wmma_f32_16x16x4_f32(
                    false, a0, false, bf, (short)0, acc[0][nt], false, false);
                acc[1][nt] = __builtin_amdgcn_wmma_f32_16x16x4_f32(
                    false, a1, false, bf, (short)0, acc[1][nt], false, false);
            }
        }
        // spill D fragments to LDS votes
#pragma unroll
        for (int mt = 0; mt < 2; ++mt)
#pragma unroll
            for (int nt = 0; nt < 5; ++nt)
#pragma unroll
                for (int v = 0; v < 8; ++v) {
                    const int row = mt * 16 + v + ((lane >> 4) << 3);
                    votes[row * VSTR + col0 + nt * 16 + ncol] = acc[mt][nt][v];
                }
        __syncthreads();

        // ---- logits_k[b,n,c] = logits_{k-1} + <votes[b,c,:], act[b,c,:]>
        for (int i = t; i < BATCH * CAPS; i += 512) {
            const int b = i / CAPS, c = i % CAPS;
            float s = first ? 0.f : logits[((long)b * NN + n) * CAPS + c];
            const float* vp = &votes[b * VSTR + c * ATOMS];
            const float* ap = &act[(b * CAPS + c) * ATOMS];
            float d = 0.f;
#pragma unroll 8
            for (int a = 0; a < ATOMS; ++a) d += vp[a] * ap[a];
            s += d;
            logits[((long)b * NN + n) * CAPS + c] = s;
            larr[i] = s;
        }
        __syncthreads();

        // ---- softmax over c (20) per batch row
        if (t < BATCH) {
            float* L = &larr[t * CAPS];
            float m = L[0];
            for (int c = 1; c < CAPS; ++c) m = fmaxf(m, L[c]);
            float sum = 0.f;
            for (int c = 0; c < CAPS; ++c) { const float e = __expf(L[c] - m); L[c] = e; sum += e; }
            const float inv = 1.f / sum;
            for (int c = 0; c < CAPS; ++c) L[c] *= inv;
        }
        __syncthreads();

        // ---- pacc[b,ca] += route[b,c] * votes[b,ca]
#pragma unroll
        for (int j = 0; j < 80; ++j) {
            const int idx = j * 512 + t;          // 0..40959 over [32][1280]
            const int b = idx / CA;
            const int ca = idx - b * CA;
            pacc[j] += larr[b * CAPS + (ca >> 6)] * votes[b * VSTR + ca];
        }
    }

#pragma unroll
    for (int j = 0; j < 80; ++j) atomicAdd(&preact[j * 512 + t], pacc[j]);
}

// -------------------------------------------------------------------------
// Squash epilogue: act[b,c,:] = squash(preact[b,c,:] * scale + bias[c,:])
// Grid: B*C = 640 blocks x 64 threads.
// -------------------------------------------------------------------------
__global__ void capsule_squash_kernel(const float* __restrict__ preact,
                                      const float* __restrict__ bias,
                                      float* __restrict__ out, float scale)
{
    __shared__ float red[ATOMS];
    const int bc = blockIdx.x;
    const int c  = bc % CAPS;
    const int a  = threadIdx.x;
    const float v = preact[bc * ATOMS + a] * scale + bias[c * ATOMS + a];
    red[a] = v * v;
    __syncthreads();
    for (int s = 32; s > 0; s >>= 1) {
        if (a < s) red[a] += red[a + s];
        __syncthreads();
    }
    const float ns = red[0];
    out[bc * ATOMS + a] = v * sqrtf(ns) / (1.f + ns);   // v/|v| * ns/(1+ns)
}

// -------------------------------------------------------------------------
extern "C" void kernel_launch(void* const* d_in, const int* in_sizes, int n_in,
                              void* d_out, int out_size, void* d_ws, size_t ws_size,
                              hipStream_t stream)
{
    const float* x  = (const float*)d_in[0];   // [32,10,10,32,32] == [B,N,IA]
    const float* W  = (const float*)d_in[1];   // [N, IA, C*A]
    const float* bb = (const float*)d_in[2];   // [C, A]
    float* out = (float*)d_out;                // [B, C, A]

    float* preact = (float*)d_ws;              // 40960 f
    float* actbuf = preact + BATCH * CA;       // 40960 f
    float* logits = actbuf + BATCH * CA;       // 2,048,000 f

    const size_t ldsA = (size_t)BATCH * KCH * sizeof(float);                 // 128 KB
    const size_t ldsB = (size_t)(BATCH * VSTR + BATCH * IA + BATCH * CAPS) * sizeof(float);

    // iter 0: uniform route -> single big GEMM, then squash with scale 1/C
    hipMemsetAsync(preact, 0, BATCH * CA * sizeof(float), stream);
    capsule_votes_sum_kernel<<<KTOT / KCH, 512, ldsA, stream>>>(x, W, preact);
    capsule_squash_kernel<<<BATCH * CAPS, ATOMS, 0, stream>>>(preact, bb, actbuf, 1.0f / CAPS);

    // iter 1: logits1 = votes.act0 ; route = softmax ; preact1
    hipMemsetAsync(preact, 0, BATCH * CA * sizeof(float), stream);
    capsule_routing_pass_kernel<<<200, 512, ldsB, stream>>>(x, W, logits, actbuf, preact, 1);
    capsule_squash_kernel<<<BATCH * CAPS, ATOMS, 0, stream>>>(preact, bb, actbuf, 1.0f);

    // iter 2: logits2 = logits1 + votes.act1 ; final activation -> d_out
    hipMemsetAsync(preact, 0, BATCH * CA * sizeof(float), stream);
    capsule_routing_pass_kernel<<<200, 512, ldsB, stream>>>(x, W, logits, actbuf, preact, 0);
    capsule_squash_kernel<<<BATCH * CAPS, ATOMS, 0, stream>>>(preact, bb, out, 1.0f);
}